// MetaLSTM_64304250355955
// MI455X (gfx1250) — compile-verified
//
#include <hip/hip_runtime.h>

typedef __attribute__((ext_vector_type(2))) float v2f;
typedef __attribute__((ext_vector_type(8))) float v8f;

#define WAVES_PER_BLOCK 8
#define ROWS_PER_WAVE   16
#define ROWS_PER_BLOCK  (WAVES_PER_BLOCK * ROWS_PER_WAVE)

// Meta-LSTM scan. Each wave owns 16 batch rows. Per timestep:
//   xWF[m] = x[t][m][:16] . WF[:16]   (WMMA D column 0)
//   xWI[m] = x[t][m][:16] . WI[:16]   (WMMA D column 1)
// via a K=16 chain of four v_wmma_f32_16x16x4_f32. The x tile (1KB, contiguous)
// is streamed global->LDS with async b128 copies, double-buffered one timestep
// ahead. The recurrence tail is per-lane VALU with fast rcp sigmoids.
__global__ __launch_bounds__(256) void metalstm_wmma_kernel(
    const float* __restrict__ x,     // [T, B, 16]
    const float* __restrict__ grads, // [T, B, 1]
    const float* __restrict__ WF,    // [18]
    const float* __restrict__ WI,    // [18]
    const float* __restrict__ cI,    // [B]
    const float* __restrict__ bI,    // [1]
    const float* __restrict__ bF,    // [1]
    const float* __restrict__ mPtr,  // [1]
    float* __restrict__ out,         // [4*B] : fS | iS | cS | dS
    int T, int B)
{
    // Per-wave D-matrix spill (1KB) + double-buffered x staging (2KB).
    __shared__ float spill[WAVES_PER_BLOCK * 256];
    __shared__ float stage[WAVES_PER_BLOCK * 512];

    const int tid   = threadIdx.x;
    const int lane  = tid & 31;
    const int wave  = tid >> 5;
    const int r     = lane & 15;          // batch row within tile (dup in upper half-wave)
    const int khalf = (lane >> 4) << 1;   // 0 or 2: K sub-offset per A/B VGPR layout
    const int b0    = blockIdx.x * ROWS_PER_BLOCK + wave * ROWS_PER_WAVE;
    const int row   = b0 + r;
    const int rowc  = (row < B) ? row : (B - 1);

    float* const sp  = &spill[wave * 256];
    float* const st0 = &stage[wave * 512];
    float* const st1 = st0 + 256;

    // ---- uniform scalar parameters ----
    const float WF16 = WF[16], WF17 = WF[17];
    const float WI16 = WI[16], WI17 = WI[17];
    const float bFv = bF[0], bIv = bI[0], mv = mPtr[0];

    // ---- four 4x16 B-matrices (one per K-step): col0 = WF chunk, col1 = WI chunk
    // B layout: lane n<16 holds (K=0,K=1) of column n; lane n+16 holds (K=2,K=3).
    v2f bmat[4];
    {
        const float* Wp = (r == 0) ? WF : WI;
#pragma unroll
        for (int s = 0; s < 4; ++s) {
            float e0 = 0.0f, e1 = 0.0f;
            if (r < 2) {
                e0 = Wp[4 * s + khalf + 0];
                e1 = Wp[4 * s + khalf + 1];
            }
            v2f bv = {e0, e1};
            bmat[s] = bv;
        }
    }

    // Async copy of one 1KB x tile (16 rows x 16 f32) into an LDS buffer.
    // Each lane moves 16B; instruction offset applies to BOTH global and LDS
    // addresses, so base + offset:512 covers the whole tile with 2 ops.
    auto issue_stage = [&](float* dst, int t) {
        unsigned long long ga = (unsigned long long)(const void*)
            (x + (((size_t)t * (size_t)B + (size_t)b0) << 4)) + (unsigned)(lane * 16);
        unsigned la = (unsigned)(unsigned long long)(void*)dst + (unsigned)(lane * 16);
        asm volatile("global_load_async_to_lds_b128 %0, %1, off"
                     :: "v"(la), "v"(ga) : "memory");
        asm volatile("global_load_async_to_lds_b128 %0, %1, off offset:512"
                     :: "v"(la), "v"(ga) : "memory");
    };

    // ---- recurrence state (lane m owns row m; upper half-wave duplicates) ----
    float fS = 0.0f, iS = 0.0f, dS = 0.0f;
    float cS = cI[rowc];

    issue_stage(st0, 0);

    for (int t = 0; t < T; ++t) {
        float* stc = (t & 1) ? st1 : st0;
        float* stn = (t & 1) ? st0 : st1;
        if (t + 1 < T) {
            issue_stage(stn, t + 1);                    // prefetch next tile
            asm volatile("s_wait_asynccnt 2" ::: "memory");  // oldest batch done
        } else {
            asm volatile("s_wait_asynccnt 0" ::: "memory");
        }

        // A-matrix fragments from staged tile: lane -> (row r, K = 4s+khalf..+1)
        const float* ap = stc + r * 16 + khalf;
        v2f a0 = *(const v2f*)(ap + 0);
        v2f a1 = *(const v2f*)(ap + 4);
        v2f a2 = *(const v2f*)(ap + 8);
        v2f a3 = *(const v2f*)(ap + 12);

        float g = grads[(size_t)t * (size_t)B + (size_t)rowc];

        // K=16 chain: D[m][0] = x.WF, D[m][1] = x.WI
        v8f c = {};
        c = __builtin_amdgcn_wmma_f32_16x16x4_f32(false, a0, false, bmat[0],
                                                  (short)0, c, false, false);
        c = __builtin_amdgcn_wmma_f32_16x16x4_f32(false, a1, false, bmat[1],
                                                  (short)0, c, false, false);
        c = __builtin_amdgcn_wmma_f32_16x16x4_f32(false, a2, false, bmat[2],
                                                  (short)0, c, false, false);
        c = __builtin_amdgcn_wmma_f32_16x16x4_f32(false, a3, false, bmat[3],
                                                  (short)0, c, false, false);

        // Spill D to LDS, gather columns 0/1 into lane-owns-row layout.
        // D[m][n]: m<8 -> VGPR m, lane n ; m>=8 -> VGPR m-8, lane 16+n.
        // LDS float index = src_lane*8 + vgpr.
#pragma unroll
        for (int v = 0; v < 8; ++v)
            sp[lane * 8 + v] = c[v];
        asm volatile("s_wait_dscnt 0" ::: "memory");
        float xWF = sp[(r < 8) ? r       : (120 + r)];  // n=0
        float xWI = sp[(r < 8) ? (8 + r) : (128 + r)];  // n=1

        // ---- sequential recurrence tail ----
        float f_lin = xWF + cS * WF16 + fS * WF17 + bFv;
        float i_lin = xWI + cS * WI16 + iS * WI17 + bIv;
        float sf = __builtin_amdgcn_rcpf(1.0f + __expf(-f_lin));
        float si = __builtin_amdgcn_rcpf(1.0f + __expf(-i_lin));
        float d_new = mv * dS - si * g;
        float c_new = sf * cS + d_new;
        fS = f_lin; iS = i_lin; dS = d_new; cS = c_new;
    }

    if (lane < 16 && row < B) {
        out[(size_t)0 * B + row] = fS;
        out[(size_t)1 * B + row] = iS;
        out[(size_t)2 * B + row] = cS;
        out[(size_t)3 * B + row] = dS;
    }
}

extern "C" void kernel_launch(void* const* d_in, const int* in_sizes, int n_in,
                              void* d_out, int out_size, void* d_ws, size_t ws_size,
                              hipStream_t stream) {
    const float* x     = (const float*)d_in[0];
    const float* grads = (const float*)d_in[1];
    const float* WF    = (const float*)d_in[2];
    const float* WI    = (const float*)d_in[3];
    const float* cI    = (const float*)d_in[4];
    const float* bI    = (const float*)d_in[5];
    const float* bF    = (const float*)d_in[6];
    const float* m     = (const float*)d_in[7];
    float* out = (float*)d_out;

    const int B = in_sizes[4];           // cI has B elements
    const int T = in_sizes[1] / B;       // grads is T*B

    dim3 block(256);
    dim3 grid((B + ROWS_PER_BLOCK - 1) / ROWS_PER_BLOCK);
    metalstm_wmma_kernel<<<grid, block, 0, stream>>>(x, grads, WF, WI, cI, bI, bF, m,
                                                     out, T, B);
}